// ClustGeoNodeEncoder_15169824489855
// MI455X (gfx1250) — compile-verified
//
#include <hip/hip_runtime.h>

#define TILE_VOX   1000
#define TILE_ELEMS (TILE_VOX * 6)
#define ACC_STRIDE 16
#define BLOCK      256

typedef unsigned int v4u __attribute__((ext_vector_type(4)));
typedef int          v8i __attribute__((ext_vector_type(8)));
typedef int          v4i __attribute__((ext_vector_type(4)));

// ---------------------------------------------------------------------------
// Tensor Data Mover: 1D contiguous f32 tile  global -> LDS.
// D# group0: count=1 | lds_addr | global_addr lo | global_addr hi + type=2
// D# group1: data_size=4B | tensor_dim0=nElems | tensor_dim1=1 |
//            tile_dim0=nElems | tile_dim1=1 | tensor_dim0_stride=nElems
// Issued by one wave (EXEC ignored by TDM); completion via s_wait_tensorcnt.
// ---------------------------------------------------------------------------
__device__ inline void tdm_load_f32(const void* gsrc, unsigned ldsOff, unsigned nElems)
{
    unsigned long long ga = (unsigned long long)(size_t)gsrc;
    v4u g0;
    g0.x = 1u;                                                  // count = 1
    g0.y = ldsOff;                                              // lds_addr (bytes)
    g0.z = (unsigned)(ga & 0xFFFFFFFFull);                      // global_addr[31:0]
    g0.w = (unsigned)((ga >> 32) & 0x1FFFFFFull) | 0x80000000u; // [56:32] | type=2
    v8i g1;
    g1[0] = (int)(2u << 16);                       // data_size = 4 bytes
    g1[1] = (int)((nElems & 0xFFFFu) << 16);       // tensor_dim0[15:0]
    g1[2] = (int)((nElems >> 16) | (1u << 16));    // tensor_dim0[31:16] | tensor_dim1=1
    g1[3] = (int)((nElems & 0xFFFFu) << 16);       // tile_dim0 (<= 6000, fits 16b)
    g1[4] = 1;                                     // tile_dim1 = 1, tile_dim2 = 0
    g1[5] = (int)nElems;                           // tensor_dim0_stride[31:0]
    g1[6] = 0;
    g1[7] = 0;
    v4i z4 = {0, 0, 0, 0};
#if defined(__clang_major__) && (__clang_major__ >= 23)
    v8i z8 = {0, 0, 0, 0, 0, 0, 0, 0};
    __builtin_amdgcn_tensor_load_to_lds(g0, g1, z4, z4, z8, 0);
#else
    __builtin_amdgcn_tensor_load_to_lds(g0, g1, z4, z4, 0);
#endif
    __builtin_amdgcn_s_wait_tensorcnt(0);
}

// ---------------------------------------------------------------------------
// Kernel 1: zero the L2-resident accumulators (acc: nc*16 floats, sc: nc).
// ---------------------------------------------------------------------------
__global__ __launch_bounds__(BLOCK) void k_zero(float* __restrict__ acc,
                                                float* __restrict__ sc, int nc)
{
    int i = blockIdx.x * BLOCK + threadIdx.x;
    if (i < nc * ACC_STRIDE) acc[i] = 0.0f;
    if (i < nc)              sc[i]  = 0.0f;
}

// ---------------------------------------------------------------------------
// Kernel 2: fused moments pass. TDM-stage a voxel tile into LDS, then
// scatter {1, x, y, z, xx, xy, xz, yy, yz, zz} with f32 L2 atomics.
// ---------------------------------------------------------------------------
__global__ __launch_bounds__(BLOCK) void k_moments(const float* __restrict__ data,
                                                   const int*   __restrict__ cid,
                                                   float*       __restrict__ acc,
                                                   int nvox)
{
    __shared__ float tile[TILE_ELEMS];
    int base = blockIdx.x * TILE_VOX;
    int rem  = nvox - base; if (rem > TILE_VOX) rem = TILE_VOX;
    if (threadIdx.x == 0)
        tdm_load_f32(data + (size_t)base * 6,
                     (unsigned)(size_t)&tile[0], (unsigned)(rem * 6));
    __syncthreads();
    for (int v = threadIdx.x; v < rem; v += BLOCK) {
        int   c = cid[base + v];
        float x = tile[v * 6 + 1];
        float y = tile[v * 6 + 2];
        float z = tile[v * 6 + 3];
        float* a = acc + (size_t)c * ACC_STRIDE;
        unsafeAtomicAdd(a + 0, 1.0f);
        unsafeAtomicAdd(a + 1, x);
        unsafeAtomicAdd(a + 2, y);
        unsafeAtomicAdd(a + 3, z);
        unsafeAtomicAdd(a + 4, x * x);
        unsafeAtomicAdd(a + 5, x * y);
        unsafeAtomicAdd(a + 6, x * z);
        unsafeAtomicAdd(a + 7, y * y);
        unsafeAtomicAdd(a + 8, y * z);
        unsafeAtomicAdd(a + 9, z * z);
    }
}

// ---------------------------------------------------------------------------
// Kernel 3: per-cluster 3x3 symmetric eigensolve (f64 VALU, analytic),
// write center, B, size into feats; stash v0 + masked dirwt for pass 4/5.
// ---------------------------------------------------------------------------
__global__ __launch_bounds__(BLOCK) void k_eig(const float* __restrict__ acc,
                                               float* __restrict__ out,
                                               float* __restrict__ v0w,
                                               float* __restrict__ dww,
                                               int nc)
{
    int c = blockIdx.x * BLOCK + threadIdx.x;
    if (c >= nc) return;
    const float* a = acc + (size_t)c * ACC_STRIDE;
    float n     = a[0];
    float denom = fmaxf(n, 1.0f);
    float cx = a[1] / denom, cy = a[2] / denom, cz = a[3] / denom;

    double nd  = (double)n;
    double A00 = (double)a[4] - nd * (double)cx * (double)cx;
    double A01 = (double)a[5] - nd * (double)cx * (double)cy;
    double A02 = (double)a[6] - nd * (double)cx * (double)cz;
    double A11 = (double)a[7] - nd * (double)cy * (double)cy;
    double A12 = (double)a[8] - nd * (double)cy * (double)cz;
    double A22 = (double)a[9] - nd * (double)cz * (double)cz;

    double w1, w2, vx, vy, vz;
    double q  = (A00 + A11 + A22) / 3.0;
    double b00 = A00 - q, b11 = A11 - q, b22 = A22 - q;
    double p1 = A01 * A01 + A02 * A02 + A12 * A12;
    double p2 = b00 * b00 + b11 * b11 + b22 * b22 + 2.0 * p1;
    double p  = sqrt(p2 / 6.0);
    if (p2 <= 0.0 || p < 1e-300) {
        w1 = w2 = q; vx = 0.0; vy = 0.0; vz = 1.0;      // A == q*I
    } else {
        double ip  = 1.0 / p;
        double c00 = b00 * ip, c01 = A01 * ip, c02 = A02 * ip;
        double c11 = b11 * ip, c12 = A12 * ip, c22 = b22 * ip;
        double det = c00 * (c11 * c22 - c12 * c12)
                   - c01 * (c01 * c22 - c12 * c02)
                   + c02 * (c01 * c12 - c11 * c02);
        double r = 0.5 * det;
        r = fmin(1.0, fmax(-1.0, r));
        double phi = acos(r) / 3.0;
        w2 = q + 2.0 * p * cos(phi);                      // largest
        double w0 = q + 2.0 * p * cos(phi + 2.0943951023931953); // smallest
        w1 = 3.0 * q - w0 - w2;                           // middle
        // eigenvector of largest eigenvalue: cross products of rows of A - w2*I
        double m00 = A00 - w2, m11 = A11 - w2, m22 = A22 - w2;
        double c0x = A01 * A12 - A02 * m11, c0y = A02 * A01 - m00 * A12, c0z = m00 * m11 - A01 * A01;
        double c1x = A01 * m22 - A02 * A12, c1y = A02 * A02 - m00 * m22, c1z = m00 * A12 - A01 * A02;
        double c2x = m11 * m22 - A12 * A12, c2y = A12 * A02 - A01 * m22, c2z = A01 * A12 - m11 * A02;
        double n0 = c0x * c0x + c0y * c0y + c0z * c0z;
        double n1 = c1x * c1x + c1y * c1y + c1z * c1z;
        double n2 = c2x * c2x + c2y * c2y + c2z * c2z;
        double bx = c0x, by = c0y, bz = c0z, bn = n0;
        if (n1 > bn) { bx = c1x; by = c1y; bz = c1z; bn = n1; }
        if (n2 > bn) { bx = c2x; by = c2y; bz = c2z; bn = n2; }
        if (bn > 1e-300) {
            double inv = 1.0 / sqrt(bn);
            vx = bx * inv; vy = by * inv; vz = bz * inv;
        } else {                                          // degenerate top pair
            vx = 0.0; vy = 0.0; vz = 1.0;
        }
    }
    double safe  = (w2 > 0.0) ? w2 : 1.0;
    float  dirwt = (float)(1.0 - w1 / safe);
    bool   multi = (n >= 2.0f);
    double invs  = multi ? (1.0 / safe) : 0.0;

    float* f = out + (size_t)c * 16;
    f[0] = cx; f[1] = cy; f[2] = cz;
    float B00 = (float)(A00 * invs), B01 = (float)(A01 * invs), B02 = (float)(A02 * invs);
    float B11 = (float)(A11 * invs), B12 = (float)(A12 * invs), B22 = (float)(A22 * invs);
    f[3] = B00; f[4]  = B01; f[5]  = B02;
    f[6] = B01; f[7]  = B11; f[8]  = B12;
    f[9] = B02; f[10] = B12; f[11] = B22;
    f[15] = n;
    v0w[3 * c + 0] = (float)vx;
    v0w[3 * c + 1] = (float)vy;
    v0w[3 * c + 2] = (float)vz;
    dww[c] = multi ? dirwt : 0.0f;
}

// ---------------------------------------------------------------------------
// Kernel 4: orientation pass. TDM-stage tile, accumulate sc = sum(x0 * |x_perp|).
// ---------------------------------------------------------------------------
__global__ __launch_bounds__(BLOCK) void k_sc(const float* __restrict__ data,
                                              const int*   __restrict__ cid,
                                              const float* __restrict__ acc,
                                              const float* __restrict__ v0w,
                                              float*       __restrict__ sc,
                                              int nvox)
{
    __shared__ float tile[TILE_ELEMS];
    int base = blockIdx.x * TILE_VOX;
    int rem  = nvox - base; if (rem > TILE_VOX) rem = TILE_VOX;
    if (threadIdx.x == 0)
        tdm_load_f32(data + (size_t)base * 6,
                     (unsigned)(size_t)&tile[0], (unsigned)(rem * 6));
    __syncthreads();
    for (int v = threadIdx.x; v < rem; v += BLOCK) {
        int c = cid[base + v];
        const float* a = acc + (size_t)c * ACC_STRIDE;
        float n = a[0]; float denom = fmaxf(n, 1.0f);
        float cx = a[1] / denom, cy = a[2] / denom, cz = a[3] / denom;
        float vx = v0w[3 * c], vy = v0w[3 * c + 1], vz = v0w[3 * c + 2];
        float x = tile[v * 6 + 1] - cx;
        float y = tile[v * 6 + 2] - cy;
        float z = tile[v * 6 + 3] - cz;
        float x0 = x * vx + y * vy + z * vz;
        float px = x - x0 * vx, py = y - x0 * vy, pz = z - x0 * vz;
        float np0 = sqrtf(px * px + py * py + pz * pz);
        unsafeAtomicAdd(&sc[c], x0 * np0);
    }
}

// ---------------------------------------------------------------------------
// Kernel 5: finalize oriented, dirwt-scaled principal axis.
// ---------------------------------------------------------------------------
__global__ __launch_bounds__(BLOCK) void k_fin(const float* __restrict__ sc,
                                               const float* __restrict__ v0w,
                                               const float* __restrict__ dww,
                                               float* __restrict__ out,
                                               int nc)
{
    int c = blockIdx.x * BLOCK + threadIdx.x;
    if (c >= nc) return;
    float s   = sc[c];
    float sgn = (s < 0.0f) ? -1.0f : 1.0f;
    float dw  = dww[c] * sgn;     // dww already masked by (size >= 2)
    float* f  = out + (size_t)c * 16;
    f[12] = v0w[3 * c + 0] * dw;
    f[13] = v0w[3 * c + 1] * dw;
    f[14] = v0w[3 * c + 2] * dw;
}

extern "C" void kernel_launch(void* const* d_in, const int* in_sizes, int n_in,
                              void* d_out, int out_size, void* d_ws, size_t ws_size,
                              hipStream_t stream)
{
    const float* data = (const float*)d_in[0];
    const int*   cid  = (const int*)d_in[1];
    int nvox = in_sizes[1];
    int nc   = out_size / 16;
    float* out = (float*)d_out;

    // workspace layout (floats): acc[nc*16] | sc[nc] | v0w[nc*3] | dww[nc]
    float* acc = (float*)d_ws;
    float* sc  = acc + (size_t)nc * ACC_STRIDE;
    float* v0w = sc  + (size_t)nc;
    float* dww = v0w + (size_t)nc * 3;

    int zgrid = (nc * ACC_STRIDE + BLOCK - 1) / BLOCK;
    int cgrid = (nc + BLOCK - 1) / BLOCK;
    int tiles = (nvox + TILE_VOX - 1) / TILE_VOX;

    k_zero   <<<zgrid, BLOCK, 0, stream>>>(acc, sc, nc);
    k_moments<<<tiles, BLOCK, 0, stream>>>(data, cid, acc, nvox);
    k_eig    <<<cgrid, BLOCK, 0, stream>>>(acc, out, v0w, dww, nc);
    k_sc     <<<tiles, BLOCK, 0, stream>>>(data, cid, acc, v0w, sc, nvox);
    k_fin    <<<cgrid, BLOCK, 0, stream>>>(sc, v0w, dww, out, nc);
}